// BasicBlock_83683142795457
// MI455X (gfx1250) — compile-verified
//
#include <hip/hip_runtime.h>
#include <math.h>

// ---------------------------------------------------------------------------
// AdderNet BasicBlock: adder2d(3x3, pad 1) -> BN(train stats) -> ReLU
// x: [16,64,32,32] f32, w: [64,64,3,3] f32, gamma/beta: [64] f32
// out: [16,64,32,32] f32
//
// Conv loop is VALU-bound (|x-w| is not bilinear -> matrix pipes can't carry
// it). CDNA5 contributions:
//   - GLOBAL_LOAD_ASYNC_TO_LDS_B32 double-buffered staging (ASYNCcnt) so all
//     global traffic overlaps the VALU loop.
//   - BN channel statistics via V_WMMA_F32_16X16X4_F32 (ones-vector matmul).
// ---------------------------------------------------------------------------

typedef float v2f __attribute__((ext_vector_type(2)));
typedef float v8f __attribute__((ext_vector_type(8)));

typedef __attribute__((address_space(1))) int as1_int;
typedef __attribute__((address_space(3))) int as3_int;

#define C_ALL 64
#define HW    32
#define MCNT  16384.0f      // N*H*W elements per channel

// ---- CDNA5 async global->LDS copy (ASYNCcnt) with graceful fallback -------
#if defined(__AMDGCN__) && __has_builtin(__builtin_amdgcn_global_load_async_to_lds_b32)
#define HAVE_ASYNC 1
#else
#define HAVE_ASYNC 0
#endif

__device__ __forceinline__ void async_copy_b32(const float* g, float* l) {
#if HAVE_ASYNC
    __builtin_amdgcn_global_load_async_to_lds_b32(
        (as1_int*)g, (as3_int*)l, 0 /*offset*/, 0 /*cpol*/);
#else
    *l = *g;
#endif
}

#if HAVE_ASYNC
#if __has_builtin(__builtin_amdgcn_s_wait_asynccnt)
#define WAIT_ASYNC(n) __builtin_amdgcn_s_wait_asynccnt(n)
#else
#define WAIT_ASYNC(n) asm volatile("s_wait_asynccnt %0" ::"i"(n) : "memory")
#endif
#else
#define WAIT_ASYNC(n) ((void)0)
#endif

// ---------------------------------------------------------------------------
// Kernel 0: zero the stats scratch (128 floats: sum[64], sumsq[64])
// ---------------------------------------------------------------------------
__global__ void zero_ws_kernel(float* __restrict__ ws) {
    if (threadIdx.x < 128) ws[threadIdx.x] = 0.0f;
}

// ---------------------------------------------------------------------------
// Kernel 1: adder conv. Grid (n=16, cog=4, rowtile=4), block 256 (8 waves).
// Each block: one image n, 16 output channels, 8 output rows (full width).
// Each thread: 4 co x 4 px accumulators. ci staged through double-buffered
// LDS in chunks of 16 via async-to-LDS; per-wave issue count is a uniform 29
// instructions per chunk (20 x-rows of 32 dwords + 9 weight dwords/lane), so
// s_wait_asynccnt 29 retires exactly the previous chunk.
// ---------------------------------------------------------------------------
__global__ __launch_bounds__(256) void adder_conv_kernel(
        const float* __restrict__ x, const float* __restrict__ w,
        float* __restrict__ out) {
    __shared__ float xs[2][16][10][35];  // [buf][ci][row+halo][col+halo, pad->35]
    __shared__ float wl[2][2304];        // [buf][co*144 + ci*9 + tap]
    __shared__ float dummy[32];          // sink for OOB halo rows

    const int n    = blockIdx.x;
    const int cog  = blockIdx.y;            // 16-channel co group
    const int rt   = blockIdx.z;            // 8-row tile
    const int tid  = threadIdx.x;
    const int lane = tid & 31;
    const int wave = tid >> 5;
    const int co4  = tid >> 6;              // 0..3   (4 co per thread)
    const int pix  = tid & 63;
    const int row  = pix >> 3;              // 0..7
    const int col4 = (pix & 7) << 2;        // 0,4,...,28
    const int row0 = rt * 8;

    // Zero both buffers once: halo columns/rows stay zero forever (= padding);
    // interior cells are rewritten by every chunk's async loads.
    {
        float* p = &xs[0][0][0][0];
        for (int i = tid; i < 2 * 16 * 10 * 35; i += 256) p[i] = 0.0f;
        if (tid < 32) dummy[tid] = 0.0f;
    }
    __syncthreads();

    // ---- per-chunk async issue: uniform 29 instructions per wave ----
    auto issue_chunk = [&](int cc, int buf) {
        // x: wave owns 20 tile rows (rowid = ci*10 + r), lane = column 0..31
        const int rbase = wave * 20;
        #pragma unroll
        for (int i = 0; i < 20; ++i) {
            const int rowid = rbase + i;
            const int ci    = rowid / 10;
            const int r     = rowid - ci * 10;
            const int gr    = row0 + r - 1;
            const bool oob  = ((unsigned)gr >= 32u);
            const int  grc  = oob ? 0 : gr;
            const float* gp = x + ((n * C_ALL + cc * 16 + ci) * HW + grc) * HW + lane;
            float* lp = oob ? &dummy[lane] : &xs[buf][ci][r][1 + lane];
            async_copy_b32(gp, lp);
        }
        // weights: wave owns 288 dwords, 9 per lane
        const int wbase = wave * 288;
        #pragma unroll
        for (int i = 0; i < 9; ++i) {
            const int d   = wbase + i * 32 + lane;      // 0..2303
            const int co  = d / 144;
            const int rem = d - co * 144;               // ci*9 + tap
            const float* gp = w + ((cog * 16 + co) * C_ALL + cc * 16) * 9 + rem;
            async_copy_b32(gp, &wl[buf][d]);
        }
    };

    float acc[16];
    #pragma unroll
    for (int i = 0; i < 16; ++i) acc[i] = 0.0f;

    issue_chunk(0, 0);                       // prologue

    for (int cc = 0; cc < 4; ++cc) {         // ci chunks of 16
        const int buf = cc & 1;
        if (cc < 3) {
            issue_chunk(cc + 1, buf ^ 1);
            WAIT_ASYNC(29);                  // chunk cc fully landed
        } else {
            WAIT_ASYNC(0);
        }
        __syncthreads();                     // all waves' data visible

        const float(*xsb)[10][35] = xs[buf];
        const float* wlb = wl[buf];

        for (int ci = 0; ci < 16; ++ci) {
            #pragma unroll
            for (int kh = 0; kh < 3; ++kh) {
                float xv[6];
                #pragma unroll
                for (int j = 0; j < 6; ++j)
                    xv[j] = xsb[ci][row + kh][col4 + j];
                #pragma unroll
                for (int co = 0; co < 4; ++co) {
                    const float* wp = &wlb[(co4 * 4 + co) * 144 + ci * 9 + kh * 3];
                    const float w0 = wp[0], w1 = wp[1], w2 = wp[2];
                    #pragma unroll
                    for (int p = 0; p < 4; ++p) {
                        acc[co * 4 + p] += fabsf(xv[p]     - w0);
                        acc[co * 4 + p] += fabsf(xv[p + 1] - w1);
                        acc[co * 4 + p] += fabsf(xv[p + 2] - w2);
                    }
                }
            }
        }
        __syncthreads();                     // protect buf^1 before next issue
    }

    const int R = row0 + row;
    #pragma unroll
    for (int co = 0; co < 4; ++co) {
        const int c_out = cog * 16 + co4 * 4 + co;
        float4 v = make_float4(-acc[co * 4 + 0], -acc[co * 4 + 1],
                               -acc[co * 4 + 2], -acc[co * 4 + 3]);
        *(float4*)&out[((n * C_ALL + c_out) * HW + R) * HW + col4] = v;
    }
}

// ---------------------------------------------------------------------------
// Kernel 2: per-channel sum / sumsq via WMMA.
// A = 16 channels x 4 consecutive values (f32 16x4 A layout), B = ones 4x16.
// D[m,*] accumulates row (=channel) sums. 64 blocks x 8 waves = 512 waves;
// each wave owns one 16-channel group (g = wave%4) and 32 K-chunks.
// ---------------------------------------------------------------------------
__global__ __launch_bounds__(256) void stats_kernel(
        const float* __restrict__ y, float* __restrict__ ws) {
    const int tid   = threadIdx.x;
    const int lane  = tid & 31;
    const int wave  = tid >> 5;
    const int gw    = blockIdx.x * 8 + wave;   // 0..511
    const int g     = gw & 3;                  // channel group (16 ch)
    const int wslot = gw >> 2;                 // 0..127
    const int m     = lane & 15;               // A row = channel-in-group
    const int half  = lane >> 4;               // K pair select
    const int c     = g * 16 + m;

    v8f accS = {};
    v8f accQ = {};
    const v2f ones = {1.0f, 1.0f};

    for (int i = 0; i < 32; ++i) {
        const int chunk = wslot * 32 + i;      // 0..4095 -> j = 0..16383
        const int j0    = chunk << 2;
        const int n     = j0 >> 10;
        const int p     = (j0 & 1023) + half * 2;
        const v2f a = *(const v2f*)&y[((n * C_ALL + c) << 10) + p];
        accS = __builtin_amdgcn_wmma_f32_16x16x4_f32(
                   false, a, false, ones, (short)0, accS, false, false);
        const v2f asq = a * a;
        accQ = __builtin_amdgcn_wmma_f32_16x16x4_f32(
                   false, asq, false, ones, (short)0, accQ, false, false);
    }

    // C layout: lane 0 holds rows 0..7 in the 8 C regs, lane 16 rows 8..15.
    if (m == 0) {
        #pragma unroll
        for (int r = 0; r < 8; ++r) {
            const int ch = g * 16 + half * 8 + r;
            atomicAdd(&ws[ch],      accS[r]);
            atomicAdd(&ws[64 + ch], accQ[r]);
        }
    }
}

// ---------------------------------------------------------------------------
// Kernel 3: in-place BN (batch stats) + ReLU over d_out. float4 per thread.
// ---------------------------------------------------------------------------
__global__ __launch_bounds__(256) void bn_relu_kernel(
        float* __restrict__ y, const float* __restrict__ ws,
        const float* __restrict__ gamma, const float* __restrict__ beta) {
    const int i4   = blockIdx.x * 256 + threadIdx.x;
    const int base = i4 * 4;
    const int c    = (base >> 10) & 63;

    const float inv   = 1.0f / MCNT;
    const float mean  = ws[c] * inv;
    const float var   = ws[64 + c] * inv - mean * mean;
    const float scale = gamma[c] * rsqrtf(var + 1e-5f);
    const float bias  = beta[c] - mean * scale;

    float4 v = *(float4*)&y[base];
    v.x = fmaxf(fmaf(v.x, scale, bias), 0.0f);
    v.y = fmaxf(fmaf(v.y, scale, bias), 0.0f);
    v.z = fmaxf(fmaf(v.z, scale, bias), 0.0f);
    v.w = fmaxf(fmaf(v.w, scale, bias), 0.0f);
    *(float4*)&y[base] = v;
}

// ---------------------------------------------------------------------------
extern "C" void kernel_launch(void* const* d_in, const int* in_sizes, int n_in,
                              void* d_out, int out_size, void* d_ws, size_t ws_size,
                              hipStream_t stream) {
    const float* x     = (const float*)d_in[0];
    const float* w     = (const float*)d_in[1];
    const float* gamma = (const float*)d_in[2];
    const float* beta  = (const float*)d_in[3];
    float* out = (float*)d_out;
    float* ws  = (float*)d_ws;

    zero_ws_kernel<<<1, 128, 0, stream>>>(ws);
    adder_conv_kernel<<<dim3(16, 4, 4), 256, 0, stream>>>(x, w, out);
    stats_kernel<<<64, 256, 0, stream>>>(out, ws);
    bn_relu_kernel<<<1024, 256, 0, stream>>>(out, ws, gamma, beta);
}